// SMINSparseMeanPool_41085657154001
// MI455X (gfx1250) — compile-verified
//
#include <hip/hip_runtime.h>
#include <stdint.h>

typedef __attribute__((ext_vector_type(2))) float v2f;
typedef __attribute__((ext_vector_type(4))) float v4f;
typedef __attribute__((ext_vector_type(8))) float v8f;

#define NCLIP 64
#define CDIV  4
#define ROWS  16                      // rows per workgroup == WMMA M
#define PSTR  65                      // 64 prefix entries + leading zero
#define TABN  (CDIV * NCLIP * NCLIP)  // 16384 (c,s,e) entries per row

// decode packed table entry: bit15 = mask, [12:6] = EE, [5:0] = SS
__device__ __forceinline__ float eval_entry(unsigned v, const float* prow) {
  if (!(v & 0x8000u)) return 0.0f;
  int ss = (int)(v & 63u);
  int ee = (int)((v >> 6) & 127u);
  float num = prow[ee] - prow[ss];
  return num * __builtin_amdgcn_rcpf((float)(ee - ss));
}

__global__ __launch_bounds__(256) void smin_sparse_meanpool(
    const float* __restrict__ x, float* __restrict__ out, int nrow) {
  __shared__ float Xs[ROWS * NCLIP];          // 4 KiB input tile
  __shared__ float Ps[ROWS * PSTR];           // prefix sums, P[r][0] = 0
  __shared__ unsigned short Tab[TABN];        // 32 KiB (mask,SS,EE) table

  const int t    = threadIdx.x;
  const int wave = t >> 5;
  const int lane = t & 31;
  const int r0   = blockIdx.x * ROWS;

  // ---- Phase 1: stage X via CDNA5 async global->LDS DMA (b128/lane) ----
  if (r0 + ROWS <= nrow) {
    unsigned ldsoff = (unsigned)(uintptr_t)(&Xs[t * 4]);
    unsigned voff   = (unsigned)(t * 16);
    const float* gbase = x + (size_t)r0 * NCLIP;
    asm volatile("global_load_async_to_lds_b128 %0, %1, %2"
                 :
                 : "v"(ldsoff), "v"(voff), "s"(gbase)
                 : "memory");
    asm volatile("s_wait_asynccnt 0" ::: "memory");
  } else {  // partial tail block: guarded scalar path
    int row = t >> 4, col = (t & 15) * 4;
    v4f val = {0.0f, 0.0f, 0.0f, 0.0f};
    if (r0 + row < nrow)
      val = *(const v4f*)(x + (size_t)(r0 + row) * NCLIP + col);
    *(v4f*)(&Xs[t * 4]) = val;
  }
  __syncthreads();

  // ---- Phase 2a (wave 0): prefix sums via V_WMMA_F32_16X16X4_F32 ----
  // P(16x64) = X(16x64) @ U, U[k][n] = (k <= n). A: 16x4 chunks of X.
  if (wave == 0) {
    const int hi = lane >> 4;   // lane half
    const int m  = lane & 15;   // A row / B,D column within tile
    v2f a[16];
#pragma unroll
    for (int kc = 0; kc < 16; ++kc) {  // A layout: V0 holds K=0/2, V1 K=1/3
      int kb = kc * 4 + hi * 2;
      a[kc].x = Xs[m * NCLIP + kb];
      a[kc].y = Xs[m * NCLIP + kb + 1];
    }
#pragma unroll
    for (int j = 0; j < 4; ++j) {      // four 16-column output tiles
      v8f acc = {};
      const int col = j * 16 + m;
      const int kcmax = 4 * j + 4;     // chunks with k > col are all-zero
      for (int kc = 0; kc < kcmax; ++kc) {
        int kb = kc * 4 + hi * 2;
        v2f b;
        b.x = (kb     <= col) ? 1.0f : 0.0f;
        b.y = (kb + 1 <= col) ? 1.0f : 0.0f;
        acc = __builtin_amdgcn_wmma_f32_16x16x4_f32(
            false, a[kc], false, b, (short)0, acc, false, false);
      }
#pragma unroll
      for (int v = 0; v < 8; ++v)      // D layout: VGPR v -> row v + 8*hi
        Ps[(v + hi * 8) * PSTR + 1 + col] = acc[v];
    }
    if (lane < 16) Ps[lane * PSTR] = 0.0f;
  } else {
    // ---- Phase 2b (waves 1..7): build the shared (mask,SS,EE) table ----
    for (int i = t - 32; i < TABN; i += 224) {
      int c = i >> 12, rem = i & 4095;
      int s = rem >> 6, e = rem & 63;
      int d = e - s;
      bool mk = false;                 // pooling mask pattern (static)
      if (d >= 0) {
        if (d <= 15)       mk = true;
        else if (d <= 31)  mk = ((d & 1) == 1) && ((s & 1) == 0);
        else               mk = (d >= 35) && ((d & 3) == 3) && ((s & 3) == 0);
      }
      unsigned short v = 0;
      if (mk) {
        int L = e + 1 - s, ss, ee;
        if (L < CDIV) { ss = s; ee = e + 1; }
        else {                         // exact floor(linspace) arithmetic
          ss = s + ((L * c) >> 2);
          ee = s + ((L * (c + 1)) >> 2);
          if (ee < ss + 1) ee = ss + 1;
        }
        v = (unsigned short)(0x8000u | ((unsigned)ee << 6) | (unsigned)ss);
      }
      Tab[i] = v;
    }
  }
  __syncthreads();

  // ---- Phase 3: streaming gather + NT float4 stores (bandwidth bound) ----
  float* outb = out + (size_t)r0 * TABN;
#pragma unroll 2
  for (int idx4 = t; idx4 < ROWS * (TABN / 4); idx4 += 256) {
    int row  = idx4 >> 12;             // 4096 float4 per source row
    int elem = (idx4 & 4095) * 4;      // = c*4096 + s*64 + e0
    if (r0 + row >= nrow) break;
    const float* prow = &Ps[row * PSTR];
    uint32_t p0 = *(const uint32_t*)(&Tab[elem]);
    uint32_t p1 = *(const uint32_t*)(&Tab[elem + 2]);
    v4f o;
    o.x = eval_entry(p0 & 0xFFFFu, prow);
    o.y = eval_entry(p0 >> 16,     prow);
    o.z = eval_entry(p1 & 0xFFFFu, prow);
    o.w = eval_entry(p1 >> 16,     prow);
    __builtin_nontemporal_store(o, (v4f*)(outb + (size_t)idx4 * 4));
  }
}

extern "C" void kernel_launch(void* const* d_in, const int* in_sizes, int n_in,
                              void* d_out, int out_size, void* d_ws, size_t ws_size,
                              hipStream_t stream) {
  const float* x = (const float*)d_in[0];
  float* out = (float*)d_out;
  int nrow = in_sizes[0] / NCLIP;              // B*D = 8192
  int nblocks = (nrow + ROWS - 1) / ROWS;      // 512
  smin_sparse_meanpool<<<dim3(nblocks), dim3(256), 0, stream>>>(x, out, nrow);
}